// Encoder_34763465294349
// MI455X (gfx1250) — compile-verified
//
#include <hip/hip_runtime.h>
#include <math.h>

// TreeLSTM shift-reduce encoder for MI455X (gfx1250, wave32).
// Design: batch-uniform action schedule computed on device; shift entries
// precomputed as two batched WMMA GEMMs; 255 per-step fused reduce kernels
// (tag cell -> word cell -> in-register LSTM gating) using
// V_WMMA_F32_16X16X4_F32 for f32-exact matrix math and V_TANH_F32 for the
// gating epilogue (critical path of the sequential reduce chain).

#define Bn    256
#define Ln    128
#define WDIM  300
#define Hn    256
#define TAGn  64
#define Tn    255          // 2L-1
#define NSTEP 256          // T+1
#define WVn   50

typedef __attribute__((ext_vector_type(2))) float v2f;
typedef __attribute__((ext_vector_type(8))) float v8f;

struct Sched {
  int act[NSTEP];
  int lidx[NSTEP];
  int ridx[NSTEP];
  int tcv[NSTEP];
  int wstep[Ln + 2];
  int effT;
};

// gfx1250 hardware tanh (TRANS32) -- single instruction instead of libm's
// branchy ~30-op expansion with EXEC save/restore.
__device__ __forceinline__ float tanh_fast(float x) {
#if __has_builtin(__builtin_amdgcn_tanhf)
  return __builtin_amdgcn_tanhf(x);
#else
  return tanhf(x);
#endif
}
// exact identity: sigmoid(x) = 0.5*(tanh(x/2)+1)
__device__ __forceinline__ float sigmoid_fast(float x) {
  return __builtin_fmaf(0.5f, tanh_fast(0.5f * x), 0.5f);
}

__device__ __forceinline__ v8f wmma_f32(v2f a, v2f b, v8f c) {
  // D = A(16x4,f32) * B(4x16,f32) + C(16x16,f32)
  return __builtin_amdgcn_wmma_f32_16x16x4_f32(false, a, false, b, (short)0, c, false, false);
}

// ---------------------------------------------------------------- schedule
__global__ void sched_kernel(const int* __restrict__ a, Sched* __restrict__ sc) {
  if (threadIdx.x != 0 || blockIdx.x != 0) return;
  int Q[Ln + 2];
  for (int i = 0; i < Ln + 2; ++i) Q[i] = 0;
  for (int i = 0; i < Ln + 2; ++i) sc->wstep[i] = 0;
  int qc = 1, buf = 0, tc = 0, eff = 0;
  sc->act[0] = 0; sc->lidx[0] = 0; sc->ridx[0] = 0; sc->tcv[0] = 0;
  for (int t = 1; t <= Tn; ++t) {
    int act = a[t - 1];                 // actions are batch-uniform; row 0
    int sh = (act == 1), rd = (act == 2);
    sc->act[t] = act;
    int qm1 = qc - 1; qm1 = qm1 < 0 ? 0 : (qm1 > Ln + 1 ? Ln + 1 : qm1);
    int q0  = qc;     q0  = q0  < 0 ? 0 : (q0  > Ln + 1 ? Ln + 1 : q0);
    sc->lidx[t] = Q[qm1];
    sc->ridx[t] = Q[q0];
    sc->tcv[t]  = tc < (Tn - 1) ? tc : (Tn - 1);
    if (sh) { int bw = buf < Ln + 1 ? buf : Ln + 1; sc->wstep[bw] = t; buf++; }
    if (rd) tc++;
    int qc2 = qc + sh - rd;
    qc2 = qc2 < 0 ? 0 : (qc2 > Ln + 1 ? Ln + 1 : qc2);
    if (sh || rd) { Q[qc2] = t; eff = t; }
    qc = qc2;
  }
  sc->effT = eff;
}

// ---------------------------------------------------------------- init Stag[:,0,:] = 0
__global__ void init_stag0(float* __restrict__ Stag) {
  int i = blockIdx.x * blockDim.x + threadIdx.x;
  if (i < Bn * 2 * TAGn) {
    int b = i >> 7, c = i & 127;
    Stag[((size_t)b * NSTEP) * (2 * TAGn) + c] = 0.0f;
  }
}

// ---------------------------------------------------------------- leaf = tanh(x @ W_wl + b_wl)
// M = B*L rows, K = 300, N = 512; writes straight into S[b, wstep[w], :] (and S[b,0,:] for w==0).
__global__ __launch_bounds__(256) void leaf_kernel(const float* __restrict__ x,
                                                   const float* __restrict__ Wwl,
                                                   const float* __restrict__ bwl,
                                                   const Sched* __restrict__ sc,
                                                   float* __restrict__ S) {
  __shared__ __align__(16) float Asm[16 * 308];   // pad 308 -> conflict-free b64 reads
  const int tid = threadIdx.x, lane = tid & 31, wave = tid >> 5;
  const int row0 = blockIdx.x * 16;
  for (int i = tid; i < 16 * WDIM; i += 256) {
    int m = i / WDIM, k = i % WDIM;
    Asm[m * 308 + k] = x[(size_t)(row0 + m) * WDIM + k];
  }
  __syncthreads();
  const int hi = lane >> 4, ln = lane & 15;
  for (int nt = wave; nt < 32; nt += 8) {
    v8f acc = {};
    const int colb = nt * 16 + ln;
    for (int k0 = 0; k0 < WDIM; k0 += 4) {
      v2f av = *(const v2f*)&Asm[ln * 308 + k0 + 2 * hi];
      int r0 = k0 + 2 * hi;
      v2f bv; bv[0] = Wwl[r0 * 512 + colb]; bv[1] = Wwl[(r0 + 1) * 512 + colb];
      acc = wmma_f32(av, bv, acc);
    }
    const float bias = bwl[colb];
#pragma unroll
    for (int e = 0; e < 8; ++e) {
      int m  = e + 8 * hi;
      int gr = row0 + m;
      int b = gr >> 7, w = gr & (Ln - 1);
      float val = tanh_fast(acc[e] + bias);
      int t = sc->wstep[w];
      if (t) S[((size_t)b * NSTEP + t) * 512 + colb] = val;
      if (w == 0) S[((size_t)b * NSTEP) * 512 + colb] = val;    // S[:,0] = xpad[:,0]
    }
  }
}

// ---------------------------------------------------------------- tag_sh = tanh(emb[word_tag] @ W_tl + b_tl)
// M = B*L rows (gathered from emb), K = 64, N = 128; writes Stag[b, wstep[w], :].
__global__ __launch_bounds__(256) void tagsh_kernel(const int* __restrict__ word_tag,
                                                    const float* __restrict__ emb,
                                                    const float* __restrict__ Wtl,
                                                    const float* __restrict__ btl,
                                                    const Sched* __restrict__ sc,
                                                    float* __restrict__ Stag) {
  __shared__ __align__(16) float Asm[16 * 68];
  const int tid = threadIdx.x, lane = tid & 31, wave = tid >> 5;
  const int row0 = blockIdx.x * 16;
  for (int i = tid; i < 16 * TAGn; i += 256) {
    int m = i >> 6, k = i & 63;
    int gr = row0 + m;
    int b = gr >> 7, w = gr & (Ln - 1);
    int wt = word_tag[b * Ln + w];
    Asm[m * 68 + k] = emb[wt * TAGn + k];
  }
  __syncthreads();
  const int hi = lane >> 4, ln = lane & 15;
  const int nt = wave;                       // 8 waves, 8 N-tiles of 128 cols
  v8f acc = {};
  const int colb = nt * 16 + ln;
  for (int k0 = 0; k0 < TAGn; k0 += 4) {
    v2f av = *(const v2f*)&Asm[ln * 68 + k0 + 2 * hi];
    int r0 = k0 + 2 * hi;
    v2f bv; bv[0] = Wtl[r0 * 128 + colb]; bv[1] = Wtl[(r0 + 1) * 128 + colb];
    acc = wmma_f32(av, bv, acc);
  }
  const float bias = btl[colb];
#pragma unroll
  for (int e = 0; e < 8; ++e) {
    int m  = e + 8 * hi;
    int gr = row0 + m;
    int b = gr >> 7, w = gr & (Ln - 1);
    int t = sc->wstep[w];
    if (t) Stag[((size_t)b * NSTEP + t) * 128 + colb] = tanh_fast(acc[e] + bias);
  }
}

// ---------------------------------------------------------------- fused reduce step
// One WG per 16 batch rows. Tag cell first (waves 0-3), its h feeds the word
// cell A-matrix; 5 gate accumulators per wave -> gating fully in registers.
__global__ __launch_bounds__(256) void step_kernel(const float* __restrict__ Wtt,
                                                   const float* __restrict__ btt,
                                                   const float* __restrict__ Wwt,
                                                   const float* __restrict__ bwt,
                                                   const float* __restrict__ emb,
                                                   const int* __restrict__ cons_tag,
                                                   const Sched* __restrict__ sc,
                                                   float* __restrict__ S,
                                                   float* __restrict__ Stag,
                                                   int t) {
  if (sc->act[t] != 2) return;               // shift steps precomputed; uniform exit
  __shared__ __align__(16) float Aw[16 * 580];   // [hl | hr | tag_h], stride 580 (bank-safe)
  __shared__ __align__(16) float At[16 * 196];   // [hlt | hrt | p_tag], stride 196
  const int l = sc->lidx[t], r = sc->ridx[t], tcv = sc->tcv[t];
  const int tid = threadIdx.x, lane = tid & 31, wave = tid >> 5;
  const int b0 = blockIdx.x * 16;

  // hint: pull word-cell weights toward L2/L0 while we stage A
  for (int pi = tid; pi < 576; pi += 256) __builtin_prefetch(&Wwt[(size_t)pi * 1280], 0, 1);

  for (int i = tid; i < 16 * 512; i += 256) {
    int m = i >> 9, c = i & 511;
    float v = (c < 256) ? S[((size_t)(b0 + m) * NSTEP + l) * 512 + c]
                        : S[((size_t)(b0 + m) * NSTEP + r) * 512 + (c - 256)];
    Aw[m * 580 + c] = v;
  }
  for (int i = tid; i < 16 * 192; i += 256) {
    int m = i / 192, c = i % 192;
    float v;
    if (c < 64)        v = Stag[((size_t)(b0 + m) * NSTEP + l) * 128 + c];
    else if (c < 128)  v = Stag[((size_t)(b0 + m) * NSTEP + r) * 128 + (c - 64)];
    else { int ct = cons_tag[(b0 + m) * Tn + tcv] + WVn; v = emb[ct * TAGn + (c - 128)]; }
    At[m * 196 + c] = v;
  }
  __syncthreads();

  const int hi = lane >> 4, ln = lane & 15;

  // ---- tag cell: K=192, gates 64 wide -> 4 j-tiles, waves 0-3
  if (wave < 4) {
    const int jt = wave;
    v8f acc[5] = {};
    for (int k0 = 0; k0 < 192; k0 += 4) {
      v2f av = *(const v2f*)&At[ln * 196 + k0 + 2 * hi];
      int r0 = k0 + 2 * hi;
#pragma unroll
      for (int g = 0; g < 5; ++g) {
        int col = g * 64 + jt * 16 + ln;
        v2f bv; bv[0] = Wtt[r0 * 320 + col]; bv[1] = Wtt[(r0 + 1) * 320 + col];
        acc[g] = wmma_f32(av, bv, acc[g]);
      }
    }
    const int j = jt * 16 + ln;
    const float bi_ = btt[j], bfl = btt[64 + j], bfr = btt[128 + j],
                bu = btt[192 + j], bo = btt[256 + j];
#pragma unroll
    for (int e = 0; e < 8; ++e) {
      int m = e + 8 * hi;
      size_t brow = b0 + m;
      float zi = acc[0][e] + bi_, zfl = acc[1][e] + bfl, zfr = acc[2][e] + bfr,
            zu = acc[3][e] + bu, zo = acc[4][e] + bo;
      float clt = Stag[(brow * NSTEP + l) * 128 + 64 + j];
      float crt = Stag[(brow * NSTEP + r) * 128 + 64 + j];
      float c = sigmoid_fast(zi) * tanh_fast(zu) + sigmoid_fast(zfl) * clt + sigmoid_fast(zfr) * crt;
      float h = sigmoid_fast(zo) * tanh_fast(c);
      Stag[(brow * NSTEP + t) * 128 + j]      = h;
      Stag[(brow * NSTEP + t) * 128 + 64 + j] = c;
      Aw[m * 580 + 512 + j] = h;             // e-input of word cell
    }
  }
  __syncthreads();

  // ---- word cell: K=576, gates 256 wide -> 16 j-tiles, 2 per wave
  for (int jt = wave * 2; jt < wave * 2 + 2; ++jt) {
    v8f acc[5] = {};
    for (int k0 = 0; k0 < 576; k0 += 4) {
      v2f av = *(const v2f*)&Aw[ln * 580 + k0 + 2 * hi];
      int r0 = k0 + 2 * hi;
#pragma unroll
      for (int g = 0; g < 5; ++g) {
        int col = g * 256 + jt * 16 + ln;
        v2f bv; bv[0] = Wwt[(size_t)r0 * 1280 + col]; bv[1] = Wwt[(size_t)(r0 + 1) * 1280 + col];
        acc[g] = wmma_f32(av, bv, acc[g]);
      }
    }
    const int j = jt * 16 + ln;
    const float bi_ = bwt[j], bfl = bwt[256 + j], bfr = bwt[512 + j],
                bu = bwt[768 + j], bo = bwt[1024 + j];
#pragma unroll
    for (int e = 0; e < 8; ++e) {
      int m = e + 8 * hi;
      size_t brow = b0 + m;
      float zi = acc[0][e] + bi_, zfl = acc[1][e] + bfl, zfr = acc[2][e] + bfr,
            zu = acc[3][e] + bu, zo = acc[4][e] + bo;
      float cl = S[(brow * NSTEP + l) * 512 + 256 + j];
      float cr = S[(brow * NSTEP + r) * 512 + 256 + j];
      float c = sigmoid_fast(zi) * tanh_fast(zu) + sigmoid_fast(zfl) * cl + sigmoid_fast(zfr) * cr;
      float h = sigmoid_fast(zo) * tanh_fast(c);
      S[(brow * NSTEP + t) * 512 + j]       = h;
      S[(brow * NSTEP + t) * 512 + 256 + j] = c;
    }
  }
}

// ---------------------------------------------------------------- output = S[:, effT, :H]
__global__ void out_kernel(const float* __restrict__ S, const Sched* __restrict__ sc,
                           float* __restrict__ out) {
  int i = blockIdx.x * blockDim.x + threadIdx.x;
  if (i < Bn * Hn) {
    int b = i >> 8, j = i & 255;
    out[i] = S[((size_t)b * NSTEP + sc->effT) * 512 + j];
  }
}

// ---------------------------------------------------------------- host entry
extern "C" void kernel_launch(void* const* d_in, const int* in_sizes, int n_in,
                              void* d_out, int out_size, void* d_ws, size_t ws_size,
                              hipStream_t stream) {
  (void)in_sizes; (void)n_in; (void)out_size; (void)ws_size;
  const float* x        = (const float*)d_in[0];
  const int*   a        = (const int*)d_in[2];
  const int*   word_tag = (const int*)d_in[4];
  const int*   cons_tag = (const int*)d_in[5];
  const float* Wwl = (const float*)d_in[6];
  const float* bwl = (const float*)d_in[7];
  const float* Wtl = (const float*)d_in[8];
  const float* btl = (const float*)d_in[9];
  const float* Wwt = (const float*)d_in[10];
  const float* bwt = (const float*)d_in[11];
  const float* Wtt = (const float*)d_in[12];
  const float* btt = (const float*)d_in[13];
  const float* emb = (const float*)d_in[14];
  float* out = (float*)d_out;

  // workspace: Sched | S (B,256,512) f32 | Stag (B,256,128) f32  (~168 MB)
  char* wsb = (char*)d_ws;
  Sched* sc = (Sched*)wsb;
  size_t off = (sizeof(Sched) + 255) & ~(size_t)255;
  float* S    = (float*)(wsb + off);
  float* Stag = (float*)(wsb + off + (size_t)Bn * NSTEP * 512 * sizeof(float));

  sched_kernel<<<1, 1, 0, stream>>>(a, sc);
  init_stag0<<<(Bn * 2 * TAGn + 255) / 256, 256, 0, stream>>>(Stag);
  leaf_kernel<<<Bn * Ln / 16, 256, 0, stream>>>(x, Wwl, bwl, sc, S);
  tagsh_kernel<<<Bn * Ln / 16, 256, 0, stream>>>(word_tag, emb, Wtl, btl, sc, Stag);
  for (int t = 1; t <= Tn; ++t)
    step_kernel<<<Bn / 16, 256, 0, stream>>>(Wtt, btt, Wwt, bwt, emb, cons_tag, sc, S, Stag, t);
  out_kernel<<<(Bn * Hn + 255) / 256, 256, 0, stream>>>(S, sc, out);
}